// module_div_norm_12120397709995
// MI455X (gfx1250) — compile-verified
//
#include <hip/hip_runtime.h>
#include <math.h>
#include <float.h>

// ---------------------------------------------------------------------------
// Divisive-normalization temporal filter, MI455X (gfx1250).
//
// Per pixel p (B*C*W*H = 262144 independent series), over T = 128 steps:
//   y1[t] : recurrence A = a1*A + x[t]; Bm = a1*(Bm + A); y1[t] = Bm(pre)
//   drive = y1 / max_t(y1)
//   y2[t] : c = a2*c + drive[t]; y2[t] = c(pre)
//   norm  = y2 / max_t(y2)
//   r[t]  = nan_to_num(|drive^n| / (|norm^n| + sigma));  output r[t_q]
//
// Memory-bound: 128 MB of x read once. Strategy: async-copy each block's
// (T x 64) slab of x into LDS with global_load_async_to_lds_b128 (ASYNCcnt),
// then run both scan passes out of LDS. One HBM pass total.
// ---------------------------------------------------------------------------

#define TSTEPS       128
#define PIX_PER_BLK  64   // threads per block, pixels per block (2 waves)

__global__ __launch_bounds__(PIX_PER_BLK)
void div_norm_scan_kernel(const float* __restrict__ x,
                          const float* __restrict__ tau1,
                          const float* __restrict__ tau2,
                          const float* __restrict__ sigma,
                          const float* __restrict__ nexp,
                          const int*   __restrict__ t_ptr,
                          float*       __restrict__ out,
                          int npix, int cwh)
{
    __shared__ float tile[TSTEPS * PIX_PER_BLK];   // 32 KB: tile[t*64 + lane]

    const int tid  = threadIdx.x;
    const int base = blockIdx.x * PIX_PER_BLK;     // first pixel of this block
    const int p    = base + tid;                   // this thread's pixel
    const int psafe = (p < npix) ? p : (npix - 1);

    // ------------------------------------------------------------------
    // Stage x[:, base : base+64] -> LDS using CDNA5 async tensor-class
    // copies (GVS mode: SGPR_U64 base + per-lane VGPR_I32 byte offset).
    // 2048 B128 chunks per block = 32 per thread, fully unrolled.
    // ------------------------------------------------------------------
    {
        const unsigned long long gbase = (unsigned long long)(const void*)x;
        // Low 32 bits of a generic LDS pointer == wave-relative LDS byte offset.
        const unsigned int lds0 =
            (unsigned int)(unsigned long long)(const void*)&tile[0];

        #pragma unroll
        for (int k = 0; k < (TSTEPS * PIX_PER_BLK) / (4 * PIX_PER_BLK); ++k) {
            const int chunk = k * PIX_PER_BLK + tid;     // 0 .. 2047
            const int row   = chunk >> 4;                // time index t
            const int col   = (chunk & 15) << 2;         // pixel-in-block * 4
            const unsigned int lds_off = lds0 + (unsigned int)(chunk << 4);
            const unsigned int goff =
                (unsigned int)((((long long)row * (long long)npix)
                                + base + col) * 4ll);
            asm volatile("global_load_async_to_lds_b128 %0, %1, %2 offset:0"
                         :
                         : "v"(lds_off), "v"(goff), "s"(gbase)
                         : "memory");
        }
        asm volatile("s_wait_asynccnt 0" ::: "memory");
    }
    __syncthreads();

    // Per-pixel constants ((C,W,H) arrays broadcast over B)
    const int   pc = psafe % cwh;
    const float a1 = __expf(-1.0f / tau1[pc]);
    const float a2 = __expf(-1.0f / tau2[pc]);
    const int   tq = t_ptr[0];

    // ------------------------------------------------------------------
    // Pass 1: y1 recurrence -> max_t(y1), y1[tq]
    // ------------------------------------------------------------------
    float A = 0.0f, Bm = 0.0f;
    float max1 = -INFINITY;
    float y1_t = 0.0f;
    #pragma unroll 4
    for (int t = 0; t < TSTEPS; ++t) {
        const float xt = tile[t * PIX_PER_BLK + tid];
        const float y  = Bm;                    // y1[t] (pre-update carry)
        max1 = fmaxf(max1, y);
        if (t == tq) y1_t = y;
        const float An = a1 * A + xt;
        Bm = a1 * (Bm + A);                     // uses old A
        A  = An;
    }

    // ------------------------------------------------------------------
    // Pass 2: recompute y1 from LDS, drive = y1/max1, y2 recurrence
    //         -> max_t(y2), y2[tq]
    // ------------------------------------------------------------------
    A = 0.0f; Bm = 0.0f;
    float c = 0.0f;
    float max2 = -INFINITY;
    float y2_t = 0.0f;
    #pragma unroll 4
    for (int t = 0; t < TSTEPS; ++t) {
        const float xt = tile[t * PIX_PER_BLK + tid];
        const float y1 = Bm;                    // y1[t]
        const float d  = y1 / max1;             // drive[t]
        const float y2 = c;                     // y2[t] (pre-update carry)
        max2 = fmaxf(max2, y2);
        if (t == tq) y2_t = y2;
        c = a2 * c + d;
        const float An = a1 * A + xt;
        Bm = a1 * (Bm + A);
        A  = An;
    }

    // ------------------------------------------------------------------
    // Final element-wise math for the single requested time slice.
    // ------------------------------------------------------------------
    const float drive_t = y1_t / max1;
    const float norm_t  = y2_t / max2;
    const float nn      = nexp[pc];
    const float idr     = fabsf(powf(drive_t, nn));
    const float nrm     = fabsf(powf(norm_t, nn)) + sigma[pc];
    float r = idr / nrm;

    // jnp.nan_to_num semantics: NaN -> 0, +/-inf -> +/-FLT_MAX
    if (isnan(r))      r = 0.0f;
    else if (isinf(r)) r = (r > 0.0f) ? FLT_MAX : -FLT_MAX;

    if (p < npix) out[p] = r;
}

extern "C" void kernel_launch(void* const* d_in, const int* in_sizes, int n_in,
                              void* d_out, int out_size, void* d_ws, size_t ws_size,
                              hipStream_t stream)
{
    (void)n_in; (void)d_ws; (void)ws_size;

    const float* x     = (const float*)d_in[0];   // (T,B,C,W,H) fp32
    const float* tau1  = (const float*)d_in[1];   // (C,W,H)
    const float* tau2  = (const float*)d_in[2];   // (C,W,H)
    const float* sigma = (const float*)d_in[3];   // (C,W,H)
    const float* nexp  = (const float*)d_in[4];   // (C,W,H)
    const int*   t_q   = (const int*)d_in[5];     // scalar t
    float*       out   = (float*)d_out;           // (B,C,W,H)

    const int npix = out_size;        // B*C*W*H = 262144
    const int cwh  = in_sizes[1];     // C*W*H   = 32768
    const int nblk = (npix + PIX_PER_BLK - 1) / PIX_PER_BLK;

    div_norm_scan_kernel<<<nblk, PIX_PER_BLK, 0, stream>>>(
        x, tau1, tau2, sigma, nexp, t_q, out, npix, cwh);
}